// AIMPolicy_70944269795445
// MI455X (gfx1250) — compile-verified
//
#include <hip/hip_runtime.h>
#include <math.h>

typedef float v2f __attribute__((ext_vector_type(2)));
typedef float v8f __attribute__((ext_vector_type(8)));

#define T_DIM 16
#define GRAM_BLOCKS 2048
#define GRAM_THREADS 256   // 8 waves of 32
#define OUT_BLOCKS 4096
#define OUT_THREADS 256

// ---------------------------------------------------------------------------
// Kernel 1: partial Gram matrices via V_WMMA_F32_16X16X4_F32.
// Each wave owns a grid-strided set of K-chunks (4 columns each). Per chunk a
// lane loads float2 = G[lane&15][4c + 2*(lane>=16)], which is simultaneously
// the correct A (16x4) and B (4x16 = A^T) fragment for the symmetric product.
// Four independent load->WMMA streams keep ~1 KB of reads in flight per wave;
// 16384 waves x 1 KB ~= 16 MB in flight ~= HBM latency-bandwidth product.
// ---------------------------------------------------------------------------
__global__ __launch_bounds__(GRAM_THREADS)
void gram_partial_kernel(const float* __restrict__ G,
                         float* __restrict__ partial, long D) {
    __shared__ float sg[8][256];

    const int tid  = threadIdx.x;
    const int wave = tid >> 5;
    const int lane = tid & 31;
    const int row  = lane & 15;
    const int coff = (lane >> 4) << 1;          // 0 for lanes 0-15, 2 for 16-31

    const long chunks = D >> 2;                 // K-chunks of 4 columns
    const long stride = (long)gridDim.x * (GRAM_THREADS / 32);
    long c = (long)blockIdx.x * (GRAM_THREADS / 32) + wave;

    const float* rowp = G + (long)row * D + coff;

    v8f acc0 = {};
    v8f acc1 = {};
    v8f acc2 = {};
    v8f acc3 = {};

    // 4 independent accumulator streams -> 4 loads + 4 WMMAs in flight.
    for (; c + 3 * stride < chunks; c += 4 * stride) {
        v2f a0 = *(const v2f*)(rowp + ((c             ) << 2));
        v2f a1 = *(const v2f*)(rowp + ((c +     stride) << 2));
        v2f a2 = *(const v2f*)(rowp + ((c + 2 * stride) << 2));
        v2f a3 = *(const v2f*)(rowp + ((c + 3 * stride) << 2));
        acc0 = __builtin_amdgcn_wmma_f32_16x16x4_f32(false, a0, false, a0,
                                                     (short)0, acc0, false, false);
        acc1 = __builtin_amdgcn_wmma_f32_16x16x4_f32(false, a1, false, a1,
                                                     (short)0, acc1, false, false);
        acc2 = __builtin_amdgcn_wmma_f32_16x16x4_f32(false, a2, false, a2,
                                                     (short)0, acc2, false, false);
        acc3 = __builtin_amdgcn_wmma_f32_16x16x4_f32(false, a3, false, a3,
                                                     (short)0, acc3, false, false);
    }
    // Tail: at most 3 leftover chunks.
    for (; c < chunks; c += stride) {
        v2f a0 = *(const v2f*)(rowp + (c << 2));
        acc0 = __builtin_amdgcn_wmma_f32_16x16x4_f32(false, a0, false, a0,
                                                     (short)0, acc0, false, false);
    }
#pragma unroll
    for (int r = 0; r < 8; ++r) acc0[r] = (acc0[r] + acc1[r]) + (acc2[r] + acc3[r]);

    // C/D layout: VGPR r, lanes 0-15 -> M=r N=lane; lanes 16-31 -> M=r+8 N=lane-16.
#pragma unroll
    for (int r = 0; r < 8; ++r) {
        int m = r + ((lane >> 4) << 3);
        int n = lane & 15;
        sg[wave][m * 16 + n] = acc0[r];
    }
    __syncthreads();

    float s = 0.0f;
#pragma unroll
    for (int w = 0; w < 8; ++w) s += sg[w][tid];
    partial[(long)blockIdx.x * 256 + tid] = s;
}

// ---------------------------------------------------------------------------
// Kernel 2: reduce partial Grams, compute alpha[j] = 1 - sum_i coeff[i][j].
// coeff[i][j] = sigmoid(10*(tau - cos)) * gram/(||g_j||^2) with zero diagonal.
// ---------------------------------------------------------------------------
__global__ __launch_bounds__(256)
void coeff_kernel(const float* __restrict__ partial,
                  const float* __restrict__ tau,
                  float* __restrict__ alpha, int nblocks) {
    __shared__ float gram[256];
    __shared__ float coeff[256];

    const int t = threadIdx.x;
    float s0 = 0.0f, s1 = 0.0f, s2 = 0.0f, s3 = 0.0f;
    int b = 0;
    for (; b + 3 < nblocks; b += 4) {
        s0 += partial[(long)(b    ) * 256 + t];
        s1 += partial[(long)(b + 1) * 256 + t];
        s2 += partial[(long)(b + 2) * 256 + t];
        s3 += partial[(long)(b + 3) * 256 + t];
    }
    for (; b < nblocks; ++b) s0 += partial[(long)b * 256 + t];
    gram[t] = (s0 + s1) + (s2 + s3);
    __syncthreads();

    const int i = t >> 4;
    const int j = t & 15;
    float ni   = sqrtf(gram[i * 17]) + 1e-8f;   // i*16+i
    float nj   = sqrtf(gram[j * 17]) + 1e-8f;
    float cosv = gram[t] / (ni * nj);
    float w    = 1.0f / (1.0f + expf(-10.0f * (tau[t] - cosv)));
    float proj = gram[t] / (nj * nj);
    coeff[t]   = (i == j) ? 0.0f : (w * proj);
    __syncthreads();

    if (t < 16) {
        float cs = 0.0f;
#pragma unroll
        for (int ii = 0; ii < 16; ++ii) cs += coeff[ii * 16 + t];
        alpha[t] = 1.0f - cs;
    }
}

// ---------------------------------------------------------------------------
// Kernel 3: out[d] = sum_j alpha[j] * G[j][d], float4-vectorized streaming
// (16 global_load_b128 in flight per thread).
// ---------------------------------------------------------------------------
__global__ __launch_bounds__(OUT_THREADS)
void output_kernel(const float* __restrict__ G,
                   const float* __restrict__ alpha,
                   float* __restrict__ out, long D) {
    const long N4 = D >> 2;
    float a[16];
#pragma unroll
    for (int j = 0; j < 16; ++j) a[j] = alpha[j];

    const float4* __restrict__ G4   = (const float4*)G;
    float4* __restrict__       out4 = (float4*)out;

    long idx = (long)blockIdx.x * blockDim.x + threadIdx.x;
    const long stride = (long)gridDim.x * blockDim.x;
    for (long p = idx; p < N4; p += stride) {
        float4 acc = make_float4(0.0f, 0.0f, 0.0f, 0.0f);
#pragma unroll
        for (int j = 0; j < 16; ++j) {
            float4 v = G4[(long)j * N4 + p];
            acc.x = fmaf(a[j], v.x, acc.x);
            acc.y = fmaf(a[j], v.y, acc.y);
            acc.z = fmaf(a[j], v.z, acc.z);
            acc.w = fmaf(a[j], v.w, acc.w);
        }
        out4[p] = acc;
    }
}

extern "C" void kernel_launch(void* const* d_in, const int* in_sizes, int n_in,
                              void* d_out, int out_size, void* d_ws, size_t ws_size,
                              hipStream_t stream) {
    const float* G   = (const float*)d_in[0];   // [16, D] fp32
    const float* tau = (const float*)d_in[1];   // [16, 16] fp32
    const long D = (long)in_sizes[0] / T_DIM;

    float* partial = (float*)d_ws;                          // GRAM_BLOCKS*256 floats (2 MB)
    float* alpha   = partial + (long)GRAM_BLOCKS * 256;     // 16 floats

    gram_partial_kernel<<<GRAM_BLOCKS, GRAM_THREADS, 0, stream>>>(G, partial, D);
    coeff_kernel<<<1, 256, 0, stream>>>(partial, tau, alpha, GRAM_BLOCKS);
    output_kernel<<<OUT_BLOCKS, OUT_THREADS, 0, stream>>>(G, alpha, (float*)d_out, D);
}